// TTEmbedding_36447092474248
// MI455X (gfx1250) — compile-verified
//
#include <hip/hip_runtime.h>

// CDNA5 / gfx1250, wave32. One wave per TT-embedding lookup.
// Step 1 (B-slice x C-slice -> T, 16x16 fp32) on VALU, laid out directly in
// WMMA B-operand register layout. Step 2 (A-slice x T) as a chain of 4
// V_WMMA_F32_16X16X4_F32 (K = 16 split into 4 chunks of 4), fp32-exact.

typedef __attribute__((ext_vector_type(2))) float v2f;
typedef __attribute__((ext_vector_type(4))) float v4f;
typedef __attribute__((ext_vector_type(8))) float v8f;

// Tensor-train geometry (compile-time constants from the reference)
//   core0: (1, 50, 8, 16)   -> strides: v0:128, e0:16, r1:1
//   core1: (16, 50, 4, 16)  -> strides: r1:3200, v1:64, e1:16, r2:1
//   core2: (16, 80, 4, 1)   -> strides: r2:320, v2:4, e2:1
#define C0_V0_STRIDE 128
#define C0_E0_STRIDE 16
#define C1_R1_STRIDE 3200
#define C1_V1_STRIDE 64
#define C1_E1_STRIDE 16
#define C2_R2_STRIDE 320
#define C2_V2_STRIDE 4
#define EMBED_DIM 128
#define WAVES_PER_BLOCK 8

__global__ __launch_bounds__(256) void tt_embedding_kernel(
    const int* __restrict__ indices,
    const float* __restrict__ core0,
    const float* __restrict__ core1,
    const float* __restrict__ core2,
    float* __restrict__ out,
    int n_indices)
{
    const int lane = threadIdx.x & 31;
    const int wave = threadIdx.x >> 5;
    const int i = blockIdx.x * WAVES_PER_BLOCK + wave;
    if (i >= n_indices) return;          // wave-uniform: EXEC stays all-ones

    // Decompose vocab index (row-major over (50, 50, 80))
    const int idx = indices[i];
    const int v0  = idx / 4000;
    const int rem = idx - v0 * 4000;
    const int v1  = rem / 80;
    const int v2  = rem - v1 * 80;

    const int half = lane >> 4;          // which 16-lane half of the wave
    const int mn   = lane & 15;          // M row for A-operand == N col for B-operand
    const int e1   = mn >> 2;
    const int e2   = mn & 3;

    // Preload the C column this lane needs: c[r2] = core2[r2, v2, e2]
    float c[16];
#pragma unroll
    for (int r2 = 0; r2 < 16; ++r2)
        c[r2] = core2[r2 * C2_R2_STRIDE + v2 * C2_V2_STRIDE + e2];

    // A rows 8..15 are padding; read (mn & 7) to stay in-bounds (results for
    // D rows 8..15 are discarded at the store).
    const float* a_base = core0 + v0 * C0_V0_STRIDE + (mn & 7) * C0_E0_STRIDE;
    const float* b_base = core1 + v1 * C1_V1_STRIDE + e1 * C1_E1_STRIDE;

    v8f acc = {};
#pragma unroll
    for (int k = 0; k < 4; ++k) {
        v2f Aop, Bop;
#pragma unroll
        for (int j = 0; j < 2; ++j) {
            // r-index required by the WMMA operand layout for this lane/VGPR:
            // VGPRj holds K = 4k + 2*half + j (lanes 0-15: half=0; 16-31: half=1)
            const int r = 4 * k + 2 * half + j;

            // A-operand element: core0[0, v0, e0=mn, r1=r]
            Aop[j] = a_base[r];

            // B-operand element: T[r, mn] = sum_r2 core1[r, v1, e1, r2] * c[r2]
            const float* brow = b_base + r * C1_R1_STRIDE;  // r2 contiguous, 64B aligned
            const v4f b0 = *(const v4f*)(brow + 0);
            const v4f b1 = *(const v4f*)(brow + 4);
            const v4f b2 = *(const v4f*)(brow + 8);
            const v4f b3 = *(const v4f*)(brow + 12);
            float t;
            t  = b0.x * c[0]  + b0.y * c[1]  + b0.z * c[2]  + b0.w * c[3];
            t += b1.x * c[4]  + b1.y * c[5]  + b1.z * c[6]  + b1.w * c[7];
            t += b2.x * c[8]  + b2.y * c[9]  + b2.z * c[10] + b2.w * c[11];
            t += b3.x * c[12] + b3.y * c[13] + b3.z * c[14] + b3.w * c[15];
            Bop[j] = t;
        }
        // D = A(16x4) * B(4x16) + C, fp32 -> v_wmma_f32_16x16x4_f32
        acc = __builtin_amdgcn_wmma_f32_16x16x4_f32(
            /*neg_a=*/false, Aop, /*neg_b=*/false, Bop,
            /*c_mod=*/(short)0, acc, /*reuse_a=*/false, /*reuse_b=*/false);
    }

    // D layout: VGPR j, lanes 0-15 hold (M=j, N=lane); lanes 16-31 hold M=j+8
    // (padding rows). Only the first half stores: out[i*128 + j*16 + n].
    if (half == 0) {
        float* orow = out + (size_t)i * EMBED_DIM + mn;
#pragma unroll
        for (int j = 0; j < 8; ++j)
            orow[j * 16] = acc[j];
    }
}

extern "C" void kernel_launch(void* const* d_in, const int* in_sizes, int n_in,
                              void* d_out, int out_size, void* d_ws, size_t ws_size,
                              hipStream_t stream) {
    const int*   indices = (const int*)d_in[0];
    const float* core0   = (const float*)d_in[1];
    const float* core1   = (const float*)d_in[2];
    const float* core2   = (const float*)d_in[3];
    float*       out     = (float*)d_out;

    const int n = in_sizes[0];  // 8192 indices
    const int blocks = (n + WAVES_PER_BLOCK - 1) / WAVES_PER_BLOCK;
    tt_embedding_kernel<<<blocks, 256, 0, stream>>>(indices, core0, core1, core2, out, n);
}